// SetCriterion_34986803593800
// MI455X (gfx1250) — compile-verified
//
#include <hip/hip_runtime.h>
#include <math.h>

// ---------------------------------------------------------------------------
// DETR SetCriterion loss for MI455X (gfx1250).
// Roofline: 230.6 MB of pred_class @ 23.3 TB/s => ~10us, memory bound.
// Strategy: TDM-style async global->LDS streaming (global_load_async_to_lds),
// exact 2-pass log-softmax out of LDS (one HBM read), wave32 shuffle
// reductions, deterministic 2-stage partial-sum reduction (no float atomics).
// ---------------------------------------------------------------------------

#define ROWS_PER_BLOCK 8
#define BLOCK_THREADS  256

// Low 32 bits of a generic pointer to __shared__ == wave-relative LDS offset.
__device__ __forceinline__ unsigned lds_offset(const void* p) {
    return (unsigned)(unsigned long long)p;
}

extern "C" __global__ void __launch_bounds__(BLOCK_THREADS)
detr_loss_main(const float* __restrict__ pred_class,
               const float* __restrict__ pred_bbox,
               const int*   __restrict__ gt_class,
               const float* __restrict__ gt_bbox,
               const int*   __restrict__ row_inds,
               float*       __restrict__ partials,   // [nblocks][5]
               int rows, int Cp1)
{
    extern __shared__ float smem[];                 // [8*Cp1] row data + [64] partials
    float* pb = smem + ROWS_PER_BLOCK * Cp1;

    const int tid  = threadIdx.x;
    const int lane = tid & 31;
    const int w    = tid >> 5;

    const int row0 = blockIdx.x * ROWS_PER_BLOCK;
    const int nrow = min(ROWS_PER_BLOCK, rows - row0);
    const long long nwords = (long long)nrow * (long long)Cp1;

    if (tid < 64) pb[tid] = 0.0f;

    // ---- async copy: global pred_class tile -> LDS (CDNA5 async-to-LDS) ----
    const float* gbase = pred_class + (long long)row0 * (long long)Cp1;
    const long long nchunk = nwords >> 2;           // 16-byte chunks
    for (long long c = tid; c < nchunk; c += BLOCK_THREADS) {
        unsigned           lo = lds_offset(smem + (c << 2));
        unsigned long long ga = (unsigned long long)(gbase + (c << 2));
        asm volatile("global_load_async_to_lds_b128 %0, %1, off"
                     :: "v"(lo), "v"(ga) : "memory");
    }
    for (long long e = (nchunk << 2) + tid; e < nwords; e += BLOCK_THREADS) {
        unsigned           lo = lds_offset(smem + e);
        unsigned long long ga = (unsigned long long)(gbase + e);
        asm volatile("global_load_async_to_lds_b32 %0, %1, off"
                     :: "v"(lo), "v"(ga) : "memory");
    }
    asm volatile("s_wait_asynccnt 0" ::: "memory");
    __syncthreads();

    // ---- one wave per row: exact log-softmax from LDS ----
    const int row = row0 + w;
    if (row < rows) {
        const float* r = smem + w * Cp1;

        float m = -INFINITY;
        for (int e = lane; e < Cp1; e += 32) m = fmaxf(m, r[e]);
        for (int off = 16; off > 0; off >>= 1) m = fmaxf(m, __shfl_xor(m, off, 32));

        float s = 0.0f;
        for (int e = lane; e < Cp1; e += 32) s += __expf(r[e] - m);
        for (int off = 16; off > 0; off >>= 1) s += __shfl_xor(s, off, 32);

        if (lane == 0) {
            const int ri = row_inds[row];
            const int t  = (ri >= 0) ? gt_class[ri] : (Cp1 - 1);
            const float xt  = r[t];
            const float nll = (m + __logf(s)) - xt;
            const float wt  = (t == 0) ? 0.1f : 1.0f;

            float l1 = 0.0f, gi = 0.0f, cnt = 0.0f;
            if (ri >= 0 && t != 0) {
                const float4 p = *(const float4*)(pred_bbox + (long long)row * 4);
                const float4 g = *(const float4*)(gt_bbox   + (long long)ri  * 4);
                const float ltx = fmaxf(p.x, g.x), lty = fmaxf(p.y, g.y);
                const float rbx = fminf(p.z, g.z), rby = fminf(p.w, g.w);
                const float iw  = fmaxf(rbx - ltx, 0.0f);
                const float ih  = fmaxf(rby - lty, 0.0f);
                const float inter = iw * ih;
                const float areaA = (p.z - p.x) * (p.w - p.y);
                const float areaB = (g.z - g.x) * (g.w - g.y);
                const float uni   = areaA + areaB - inter;
                const float iou   = inter / uni;
                const float ex0 = fminf(p.x, g.x), ey0 = fminf(p.y, g.y);
                const float ex1 = fmaxf(p.z, g.z), ey1 = fmaxf(p.w, g.w);
                const float ew  = fmaxf(ex1 - ex0, 0.0f);
                const float eh  = fmaxf(ey1 - ey0, 0.0f);
                const float enc = ew * eh;
                gi  = iou - (enc - uni) / enc;
                l1  = fabsf(p.x - g.x) + fabsf(p.y - g.y) +
                      fabsf(p.z - g.z) + fabsf(p.w - g.w);
                cnt = 1.0f;
            }
            pb[w * 8 + 0] = nll * wt;
            pb[w * 8 + 1] = wt;
            pb[w * 8 + 2] = l1;
            pb[w * 8 + 3] = gi;
            pb[w * 8 + 4] = cnt;
        }
    }
    __syncthreads();

    // ---- per-block partial (fixed summation order => deterministic) ----
    if (tid < 5) {
        float a = 0.0f;
        for (int i = 0; i < ROWS_PER_BLOCK; ++i) a += pb[i * 8 + tid];
        partials[(long long)blockIdx.x * 5 + tid] = a;
    }
}

extern "C" __global__ void __launch_bounds__(BLOCK_THREADS)
detr_loss_final(const float* __restrict__ partials, int nblocks,
                float* __restrict__ out)
{
    __shared__ float acc[BLOCK_THREADS][5];
    float a0 = 0.f, a1 = 0.f, a2 = 0.f, a3 = 0.f, a4 = 0.f;
    for (int i = threadIdx.x; i < nblocks; i += BLOCK_THREADS) {
        const float* p = partials + (long long)i * 5;
        a0 += p[0]; a1 += p[1]; a2 += p[2]; a3 += p[3]; a4 += p[4];
    }
    acc[threadIdx.x][0] = a0; acc[threadIdx.x][1] = a1; acc[threadIdx.x][2] = a2;
    acc[threadIdx.x][3] = a3; acc[threadIdx.x][4] = a4;
    __syncthreads();
    if (threadIdx.x == 0) {
        float s0 = 0.f, s1 = 0.f, s2 = 0.f, s3 = 0.f, s4 = 0.f;
        for (int i = 0; i < BLOCK_THREADS; ++i) {
            s0 += acc[i][0]; s1 += acc[i][1]; s2 += acc[i][2];
            s3 += acc[i][3]; s4 += acc[i][4];
        }
        const float class_loss = s0 / s1;
        const float bbox_loss  = s2 / (s4 * 4.0f);
        const float giou_loss  = 1.0f - s3 / s4;
        out[0] = 1.0f * class_loss + 5.0f * bbox_loss + 2.0f * giou_loss;
    }
}

extern "C" void kernel_launch(void* const* d_in, const int* in_sizes, int n_in,
                              void* d_out, int out_size, void* d_ws, size_t ws_size,
                              hipStream_t stream)
{
    const float* pred_class = (const float*)d_in[0];
    const float* pred_bbox  = (const float*)d_in[1];
    const int*   gt_class   = (const int*)  d_in[2];
    const float* gt_bbox    = (const float*)d_in[3];
    const int*   row_inds   = (const int*)  d_in[4];
    // d_in[5] (col_inds) not needed: row_inds carries the full matching.

    const int rows = in_sizes[4];                 // B*Q
    const int Cp1  = in_sizes[0] / rows;          // C+1
    const int nblocks = (rows + ROWS_PER_BLOCK - 1) / ROWS_PER_BLOCK;

    float* partials = (float*)d_ws;               // nblocks*5 floats
    const size_t shmem = (size_t)(ROWS_PER_BLOCK * Cp1 + 64) * sizeof(float);

    detr_loss_main<<<nblocks, BLOCK_THREADS, shmem, stream>>>(
        pred_class, pred_bbox, gt_class, gt_bbox, row_inds, partials, rows, Cp1);
    detr_loss_final<<<1, BLOCK_THREADS, 0, stream>>>(partials, nblocks, (float*)d_out);
}